// Encoder_12515534701343
// MI455X (gfx1250) — compile-verified
//
#include <hip/hip_runtime.h>
#include <hip/hip_bf16.h>

typedef __attribute__((ext_vector_type(2))) float v2f;
typedef __attribute__((ext_vector_type(8))) float v8f;

#define IN_C  128
#define OUT_C 32
#define NEG_SLOPE 0.2f

// ---------- helpers ----------
__device__ __forceinline__ unsigned f2ord(float f) {
    unsigned u = __float_as_uint(f);
    return (u & 0x80000000u) ? ~u : (u | 0x80000000u);
}
__device__ __forceinline__ float ord2f(unsigned u) {
    return (u & 0x80000000u) ? __uint_as_float(u & 0x7fffffffu) : __uint_as_float(~u);
}

__device__ __forceinline__ v8f wmma_f32_16x16x4(v2f a, v2f b, v8f c) {
    // 8 args: (neg_a, A, neg_b, B, c_mod, C, reuse_a, reuse_b)
    return __builtin_amdgcn_wmma_f32_16x16x4_f32(false, a, false, b, (short)0, c, false, false);
}

// ---------- kernel 1: h = x @ W via V_WMMA_F32_16X16X4_F32 ----------
// One wave handles a 16-row tile of x and both 16-col tiles of W (OUT_C=32).
// A 16x4 f32 layout: lanes 0-15 -> M=lane, K={k,k+1}; lanes 16-31 -> M=lane-16, K={k+2,k+3}
// B 4x16 f32 layout: lanes 0-15 -> N=lane, K={k,k+1}; lanes 16-31 -> N=lane-16, K={k+2,k+3}
// C/D 16x16 f32:     VGPR r -> M=r (lanes 0-15) / M=r+8 (lanes 16-31), N=lane%16
__global__ void gat_gemm_kernel(const float* __restrict__ x,
                                const float* __restrict__ W,
                                float* __restrict__ h, int N) {
    const int lane = threadIdx.x & 31;
    const int wave = threadIdx.x >> 5;
    const int tile = blockIdx.x * 8 + wave;
    const int ntiles = N >> 4;            // N is a multiple of 16 (100000)
    if (tile >= ntiles) return;           // uniform per-wave exit: EXEC all-1 for WMMA
    const int r0   = tile << 4;
    const int half = lane >> 4;           // 0 or 1
    const int l15  = lane & 15;

    v8f c0 = {};
    v8f c1 = {};
    const float* xrow = x + (size_t)(r0 + l15) * IN_C;

    for (int k = 0; k < IN_C; k += 4) {
        const int ka = k + half * 2;
        v2f a;
        a.x = xrow[ka];
        a.y = xrow[ka + 1];

        v2f b0, b1;
        b0.x = W[(size_t)ka * OUT_C + l15];
        b0.y = W[(size_t)(ka + 1) * OUT_C + l15];
        b1.x = W[(size_t)ka * OUT_C + 16 + l15];
        b1.y = W[(size_t)(ka + 1) * OUT_C + 16 + l15];

        c0 = wmma_f32_16x16x4(a, b0, c0);
        c1 = wmma_f32_16x16x4(a, b1, c1);
    }

    float* hout = h + (size_t)r0 * OUT_C;
    #pragma unroll
    for (int r = 0; r < 8; ++r) {
        const int row = half ? (r + 8) : r;
        hout[(size_t)row * OUT_C + l15]      = c0[r];
        hout[(size_t)row * OUT_C + 16 + l15] = c1[r];
    }
}

// ---------- kernel 2: per-node attention logits + init accumulators ----------
__global__ void gat_node_prep_kernel(const float* __restrict__ h,
                                     const float* __restrict__ att_src,
                                     const float* __restrict__ att_dst,
                                     float* __restrict__ a_src,
                                     float* __restrict__ a_dst,
                                     unsigned* __restrict__ maxord,
                                     float* __restrict__ denom,
                                     float* __restrict__ out, int N) {
    const int n = blockIdx.x * blockDim.x + threadIdx.x;
    if (n >= N) return;
    const float* hr = h + (size_t)n * OUT_C;
    float s0 = 0.f, s1 = 0.f;
    #pragma unroll
    for (int c = 0; c < OUT_C; ++c) {
        const float v = hr[c];
        s0 += v * att_src[c];
        s1 += v * att_dst[c];
    }
    a_src[n] = s0;
    a_dst[n] = s1;
    maxord[n] = 0u;          // below f2ord of any finite float; self-loop always overwrites
    denom[n] = 0.f;
    float* orow = out + (size_t)n * OUT_C;
    #pragma unroll
    for (int c = 0; c < OUT_C; ++c) orow[c] = 0.f;
}

// ---------- kernel 3: per-dst running max over edges (incl. self loops) ----------
__global__ void gat_edge_max_kernel(const int* __restrict__ ei,
                                    const float* __restrict__ a_src,
                                    const float* __restrict__ a_dst,
                                    unsigned* __restrict__ maxord,
                                    int E, int total) {
    const int i = blockIdx.x * blockDim.x + threadIdx.x;
    if (i >= total) return;
    const int s = (i < E) ? ei[i]     : (i - E);
    const int d = (i < E) ? ei[E + i] : (i - E);
    const float t = a_src[s] + a_dst[d];
    const float e = (t > 0.f) ? t : NEG_SLOPE * t;
    atomicMax(&maxord[d], f2ord(e));
}

// ---------- kernel 4: per-dst softmax denominator ----------
__global__ void gat_edge_sum_kernel(const int* __restrict__ ei,
                                    const float* __restrict__ a_src,
                                    const float* __restrict__ a_dst,
                                    const unsigned* __restrict__ maxord,
                                    float* __restrict__ denom,
                                    int E, int total) {
    const int i = blockIdx.x * blockDim.x + threadIdx.x;
    if (i >= total) return;
    const int s = (i < E) ? ei[i]     : (i - E);
    const int d = (i < E) ? ei[E + i] : (i - E);
    const float t = a_src[s] + a_dst[d];
    const float e = (t > 0.f) ? t : NEG_SLOPE * t;
    const float p = __expf(e - ord2f(maxord[d]));
    atomicAdd(&denom[d], p);
}

// ---------- kernel 5: weighted scatter-aggregate; 32 lanes = 32 channels of one edge ----------
__global__ void gat_edge_aggr_kernel(const int* __restrict__ ei,
                                     const float* __restrict__ a_src,
                                     const float* __restrict__ a_dst,
                                     const unsigned* __restrict__ maxord,
                                     const float* __restrict__ denom,
                                     const float* __restrict__ h,
                                     float* __restrict__ out,
                                     int E, int total) {
    const long long t = (long long)blockIdx.x * blockDim.x + threadIdx.x;
    const int i = (int)(t >> 5);          // edge id (same for all 32 lanes of a wave)
    const int c = (int)(t & 31);          // channel
    if (i >= total) return;
    const int s = (i < E) ? ei[i]     : (i - E);
    const int d = (i < E) ? ei[E + i] : (i - E);
    const float tt = a_src[s] + a_dst[d];
    const float e = (tt > 0.f) ? tt : NEG_SLOPE * tt;
    const float alpha = __expf(e - ord2f(maxord[d])) / denom[d];
    atomicAdd(&out[(size_t)d * OUT_C + c], alpha * h[(size_t)s * OUT_C + c]);
}

// ---------- kernel 6: bias + ReLU ----------
__global__ void gat_finalize_kernel(const float* __restrict__ bias,
                                    float* __restrict__ out, int total) {
    const int i = blockIdx.x * blockDim.x + threadIdx.x;
    if (i >= total) return;
    const float v = out[i] + bias[i & (OUT_C - 1)];
    out[i] = (v > 0.f) ? v : 0.f;
}

extern "C" void kernel_launch(void* const* d_in, const int* in_sizes, int n_in,
                              void* d_out, int out_size, void* d_ws, size_t ws_size,
                              hipStream_t stream) {
    const float* x       = (const float*)d_in[0];
    const int*   ei      = (const int*)  d_in[1];
    const float* W       = (const float*)d_in[2];
    const float* att_src = (const float*)d_in[3];
    const float* att_dst = (const float*)d_in[4];
    const float* bias    = (const float*)d_in[5];

    const int N = in_sizes[0] / IN_C;
    const int E = in_sizes[1] / 2;
    const int total_edges = E + N;

    float* out = (float*)d_out;                   // [N, 32] float, then edge_index tail

    // workspace: h[N*32] | a_src[N] | a_dst[N] | maxord[N] | denom[N]  (~14.4 MB)
    float*    h      = (float*)d_ws;
    float*    a_src  = h + (size_t)N * OUT_C;
    float*    a_dst  = a_src + N;
    unsigned* maxord = (unsigned*)(a_dst + N);
    float*    denom  = (float*)(maxord + N);

    // 1) GEMM: 8 waves/block, one 16-row tile per wave
    {
        const int ntiles = N / 16;
        const int blocks = (ntiles + 7) / 8;
        gat_gemm_kernel<<<blocks, 256, 0, stream>>>(x, W, h, N);
    }
    // 2) node logits + init
    gat_node_prep_kernel<<<(N + 255) / 256, 256, 0, stream>>>(
        h, att_src, att_dst, a_src, a_dst, maxord, denom, out, N);
    // 3) segment max
    gat_edge_max_kernel<<<(total_edges + 255) / 256, 256, 0, stream>>>(
        ei, a_src, a_dst, maxord, E, total_edges);
    // 4) segment sum of exp
    gat_edge_sum_kernel<<<(total_edges + 255) / 256, 256, 0, stream>>>(
        ei, a_src, a_dst, maxord, denom, E, total_edges);
    // 5) weighted scatter-aggregate
    {
        const long long threads = (long long)total_edges * OUT_C;
        const long long blocks = (threads + 255) / 256;
        gat_edge_aggr_kernel<<<(unsigned)blocks, 256, 0, stream>>>(
            ei, a_src, a_dst, maxord, denom, h, out, E, total_edges);
    }
    // 6) bias + ReLU
    gat_finalize_kernel<<<(N * OUT_C + 255) / 256, 256, 0, stream>>>(bias, out, N * OUT_C);

    // tuple tail: copy edge_index (2*E int32) after the [N,32] output
    hipMemcpyAsync((char*)d_out + (size_t)N * OUT_C * sizeof(float), d_in[1],
                   (size_t)2 * E * sizeof(int), hipMemcpyDeviceToDevice, stream);
}